// HeadLoss_29137058136626
// MI455X (gfx1250) — compile-verified
//
#include <hip/hip_runtime.h>
#include <math.h>

typedef __attribute__((ext_vector_type(2))) float v2f;
typedef __attribute__((ext_vector_type(8))) float v8f;

#define B_N 8
#define R_N 2000
#define G_N 64
#define C_N 16
#define TILES_PER_IMG (R_N / 16)            /* 125 */
#define TOTAL_TILES   (B_N * TILES_PER_IMG) /* 1000 */
#define WAVES_PER_BLOCK 4
#define DEG2RAD 0.017453292519943295f
#define RAD2DEG 57.29577951308232f

// Sutherland-Hodgman clip of ROI quad by GT quad (both CCW), then shoelace.
// g layout: [0..3]=gx corners, [4..7]=gy corners, [8]=area(w*h)
__device__ __forceinline__ float iou_pair(const float* rx, const float* ry,
                                          float areaR, const float* g) {
  float px[10], py[10];
  int n = 4;
#pragma unroll
  for (int k = 0; k < 4; ++k) { px[k] = rx[k]; py[k] = ry[k]; }
#pragma unroll
  for (int j = 0; j < 4; ++j) {
    float cx0 = g[j], cy0 = g[4 + j];
    float ex = g[(j + 1) & 3] - cx0, ey = g[4 + ((j + 1) & 3)] - cy0;
    float qx[10], qy[10];
    int m = 0;
    for (int k = 0; k < n; ++k) {
      int k2 = (k + 1 == n) ? 0 : k + 1;
      float sx = px[k], sy = py[k], tx = px[k2], ty = py[k2];
      float sp = ex * (sy - cy0) - ey * (sx - cx0);
      float sq = ex * (ty - cy0) - ey * (tx - cx0);
      bool ip = (sp >= 0.0f), iq = (sq >= 0.0f);
      if (ip && m < 10) { qx[m] = sx; qy[m] = sy; ++m; }
      if (ip != iq && m < 10) {
        float t = sp / (sp - sq);
        qx[m] = sx + t * (tx - sx);
        qy[m] = sy + t * (ty - sy);
        ++m;
      }
    }
    n = m;
    for (int k = 0; k < n; ++k) { px[k] = qx[k]; py[k] = qy[k]; }
    if (n == 0) break;
  }
  float s = 0.0f;
  for (int k = 0; k < n; ++k) {
    int k2 = (k + 1 == n) ? 0 : k + 1;
    s += px[k] * py[k2] - px[k2] * py[k];
  }
  float inter = 0.5f * fabsf(s);
  float u = fmaxf(areaR + g[8] - inter, 1e-9f);
  return fminf(fmaxf(inter / u, 0.0f), 1.0f);
}

// One wave per 16-ROI tile. WMMA f32 16x16x4 computes the circumradius-margin
// M = dist^2 - 0.5*(dr^2 + dg^2); M > 0 => boxes provably disjoint (QM>=AM).
__global__ __launch_bounds__(128) void iou_assign_kernel(
    const float* __restrict__ rois, const float* __restrict__ gts,
    float* __restrict__ miou, int* __restrict__ marg) {
  // per-wave GT stash: [0..3]=gx,[4..7]=gy,[8]=area,[9]=cx,[10]=cy,[11]=q,[12]=pad
  __shared__ float sh[WAVES_PER_BLOCK][G_N * 13];
  const int lane = threadIdx.x & 31;
  const int wv = threadIdx.x >> 5;
  const int tile = blockIdx.x * WAVES_PER_BLOCK + wv;
  const int b = tile / TILES_PER_IMG;
  const int r0 = (tile % TILES_PER_IMG) * 16;

  // ---- GT preprocessing (2 GTs / lane) ----
  for (int g = lane; g < G_N; g += 32) {
    const float* gp = gts + (b * G_N + g) * 5;
    float cx = gp[0], cy = gp[1], w = gp[2], h = gp[3];
    float t = -gp[4];  // tgt angle = -a*(180/pi) deg -> radians = -a
    float c = cosf(t), s = sinf(t);
    float hw = 0.5f * w, hh = 0.5f * h;
    float dx[4] = {-hw, hw, hw, -hw};
    float dy[4] = {-hh, -hh, hh, hh};
    float* o = &sh[wv][g * 13];
#pragma unroll
    for (int k = 0; k < 4; ++k) {
      o[k] = cx + dx[k] * c - dy[k] * s;
      o[4 + k] = cy + dx[k] * s + dy[k] * c;
    }
    o[8] = w * h;
    o[9] = cx;
    o[10] = cy;
    o[11] = cx * cx + cy * cy - 0.5f * (w * w + h * h);  // q_g
  }
  __syncthreads();

  // ---- ROI features (lane L and L+16 both own ROI L%16) ----
  const int rloc = lane & 15;
  const float* rp = rois + (b * R_N + r0 + rloc) * 5;
  float cx = rp[0], cy = rp[1], w = rp[2], h = rp[3];
  float t = rp[4] * DEG2RAD;
  float c = cosf(t), s = sinf(t);
  float rx[4], ry[4];
  {
    float hw = 0.5f * w, hh = 0.5f * h;
    float dx[4] = {-hw, hw, hw, -hw};
    float dy[4] = {-hh, -hh, hh, hh};
#pragma unroll
    for (int k = 0; k < 4; ++k) {
      rx[k] = cx + dx[k] * c - dy[k] * s;
      ry[k] = cy + dx[k] * s + dy[k] * c;
    }
  }
  float areaR = w * h;
  float qr = cx * cx + cy * cy - 0.5f * (w * w + h * h);

  // A (16x4 f32, ISA 7.12.2): lanes 0-15 hold K={0,1}, lanes 16-31 K={2,3}.
  // Row r = [x_r, y_r, q_r, 1]
  v2f a;
  a.x = (lane < 16) ? cx : qr;
  a.y = (lane < 16) ? cy : 1.0f;

  const int rsel = rloc & 7;           // which D-VGPR holds my roi row
  const bool hi = (rloc >= 8);         // my row lives in lanes16-31 ballot half

  unsigned long long cand = 0ull;
#pragma unroll
  for (int tt = 0; tt < 4; ++tt) {
    const float* gp = &sh[wv][(tt * 16 + rloc) * 13];
    float gx = gp[9], gy = gp[10], gq = gp[11];
    // B (4x16): rows striped across lanes within a VGPR.
    // VGPR0: lanes0-15 K=0 (-2x_g), lanes16-31 K=1 (-2y_g)
    // VGPR1: lanes0-15 K=2 (1),     lanes16-31 K=3 (q_g)
    v2f bf;
    bf.x = (lane < 16) ? (-2.0f * gx) : (-2.0f * gy);
    bf.y = (lane < 16) ? 1.0f : gq;
    v8f cacc = {0.f, 0.f, 0.f, 0.f, 0.f, 0.f, 0.f, 0.f};
    v8f M = __builtin_amdgcn_wmma_f32_16x16x4_f32(false, a, false, bf,
                                                  (short)0, cacc, false, false);
    // D layout: VGPR v -> row v (lanes 0-15) / row v+8 (lanes 16-31), N=lane%16
    // Grab my row's ballot inside the unrolled loop: branch-free cndmask per v,
    // instead of dynamically indexing a uniform array afterwards.
    unsigned mybal = 0u;
#pragma unroll
    for (int v = 0; v < 8; ++v) {
      unsigned balv = (unsigned)__ballot(M[v] <= 0.0f);
      mybal = (rsel == v) ? balv : mybal;
    }
    unsigned bits = hi ? (mybal >> 16) : (mybal & 0xFFFFu);
    cand |= ((unsigned long long)bits) << (16 * tt);
  }

  // ---- exact IoU only on survivors; lane L: gts 0..31, lane L+16: 32..63 ----
  unsigned bits32 = (lane < 16) ? (unsigned)(cand & 0xFFFFFFFFull)
                                : (unsigned)(cand >> 32);
  const int gbase = (lane < 16) ? 0 : 32;
  float best = 0.0f;
  int barg = 0;
  while (bits32) {
    int gg = gbase + (__ffs(bits32) - 1);
    bits32 &= bits32 - 1;
    float v = iou_pair(rx, ry, areaR, &sh[wv][gg * 13]);
    if (v > best) { best = v; barg = gg; }  // ascending scan => first-max
  }
  float ob = __shfl_xor(best, 16, 32);
  int oa = __shfl_xor(barg, 16, 32);
  if (lane < 16) {
    if (ob > best) { best = ob; barg = oa; }  // strict > keeps lower gt index
    miou[b * R_N + r0 + lane] = best;
    marg[b * R_N + r0 + lane] = barg;
  }
}

// One block per image: ordered pos/neg sampling + CE + smooth-L1.
__global__ __launch_bounds__(256) void loss_kernel(
    const float* __restrict__ gts, const float* __restrict__ bp,
    const float* __restrict__ cp, const int* __restrict__ gcls,
    const float* __restrict__ miou, const int* __restrict__ marg,
    float* __restrict__ per_img) {
  const int b = blockIdx.x;
  __shared__ float smi[R_N];
  __shared__ int pos_sel[128];
  __shared__ int neg_sel[256];
  __shared__ int s_np, s_nneg;
  __shared__ float red[256];

  for (int r = threadIdx.x; r < R_N; r += 256) smi[r] = miou[b * R_N + r];
  __syncthreads();

  if (threadIdx.x == 0) {
    int np = 0, nn = 0, negc = 0;
    for (int r = 0; r < R_N; ++r) {
      float mi = smi[r];
      if (mi >= 0.5f) {
        if (np < 128) pos_sel[np++] = r;
      } else {
        if (nn < 256) neg_sel[nn++] = r;
        ++negc;
      }
    }
    s_np = np;  // == min(total_pos, 128)
    int cap = 256 - np;
    s_nneg = (negc < cap) ? negc : cap;
  }
  __syncthreads();
  const int np = s_np, nneg = s_nneg;

  // cross entropy over valid slots
  float lsum = 0.0f;
  for (int sidx = threadIdx.x; sidx < np + nneg; sidx += 256) {
    int roi, tc;
    if (sidx < np) {
      roi = pos_sel[sidx];
      tc = gcls[b * G_N + marg[b * R_N + roi]];
    } else {
      roi = neg_sel[sidx - np];
      tc = C_N - 1;  // background
    }
    const float* l = cp + ((size_t)(b * R_N + roi)) * C_N;
    float lv[C_N], mx = -3.0e38f;
#pragma unroll
    for (int i = 0; i < C_N; ++i) {
      float x = l[i];
      if (x == 0.0f) x = 1e-7f;
      lv[i] = x;
      mx = fmaxf(mx, x);
    }
    float se = 0.0f;
#pragma unroll
    for (int i = 0; i < C_N; ++i) se += expf(lv[i] - mx);
    lsum += (mx + logf(se)) - lv[tc];
  }

  // smooth-L1 regression over positives
  float rsum = 0.0f;
  for (int sidx = threadIdx.x; sidx < np; sidx += 256) {
    int roi = pos_sel[sidx];
    int g = marg[b * R_N + roi];
    const float* gb = gts + (b * G_N + g) * 5;
    const float* bq = bp + ((size_t)(b * R_N + roi)) * 5;
#pragma unroll
    for (int k = 0; k < 5; ++k) {
      float tv = (k == 4) ? (-gb[4] * RAD2DEG) : gb[k];
      float d = fabsf(bq[k] - tv);
      rsum += (d < 1.0f) ? (0.5f * d * d) : (d - 0.5f);
    }
  }

  red[threadIdx.x] = lsum;
  __syncthreads();
  for (int off = 128; off > 0; off >>= 1) {
    if (threadIdx.x < off) red[threadIdx.x] += red[threadIdx.x + off];
    __syncthreads();
  }
  float totL = red[0];
  __syncthreads();
  red[threadIdx.x] = rsum;
  __syncthreads();
  for (int off = 128; off > 0; off >>= 1) {
    if (threadIdx.x < off) red[threadIdx.x] += red[threadIdx.x + off];
    __syncthreads();
  }
  if (threadIdx.x == 0) {
    int vs = np + nneg;
    float cls = totL / (float)((vs > 1) ? vs : 1);
    int dn = np * 5;
    float rg = red[0] / (float)((dn > 1) ? dn : 1);
    per_img[b * 3 + 0] = cls;
    per_img[b * 3 + 1] = rg;
    per_img[b * 3 + 2] = (np > 0) ? 1.0f : 0.0f;
  }
}

__global__ void finalize_kernel(const float* __restrict__ per_img,
                                float* __restrict__ out) {
  if (threadIdx.x == 0 && blockIdx.x == 0) {
    float cs = 0.0f, rs = 0.0f, ws = 0.0f;
    for (int b = 0; b < B_N; ++b) {
      cs += per_img[b * 3 + 0];
      rs += per_img[b * 3 + 1] * per_img[b * 3 + 2];
      ws += per_img[b * 3 + 2];
    }
    float cls = cs / (float)B_N;
    float rg = (ws > 0.0f) ? (rs / fmaxf(ws, 1.0f)) : 0.0f;
    if (cls != cls) cls = 0.0f;
    if (rg != rg) rg = 0.0f;
    out[0] = cls + rg;
    out[1] = cls;
    out[2] = rg;
  }
}

extern "C" void kernel_launch(void* const* d_in, const int* in_sizes, int n_in,
                              void* d_out, int out_size, void* d_ws,
                              size_t ws_size, hipStream_t stream) {
  const float* rois = (const float*)d_in[0];
  const float* gts = (const float*)d_in[1];
  const float* bp = (const float*)d_in[2];
  const float* cp = (const float*)d_in[3];
  const int* gcls = (const int*)d_in[4];
  float* out = (float*)d_out;

  float* miou = (float*)d_ws;                  // B_N*R_N floats
  int* marg = (int*)(miou + B_N * R_N);        // B_N*R_N ints
  float* per_img = (float*)(marg + B_N * R_N); // B_N*3 floats

  iou_assign_kernel<<<TOTAL_TILES / WAVES_PER_BLOCK, 32 * WAVES_PER_BLOCK, 0,
                      stream>>>(rois, gts, miou, marg);
  loss_kernel<<<B_N, 256, 0, stream>>>(gts, bp, cp, gcls, miou, marg, per_img);
  finalize_kernel<<<1, 32, 0, stream>>>(per_img, out);
}